// MyModel_87522843561110
// MI455X (gfx1250) — compile-verified
//
#include <hip/hip_runtime.h>

typedef __attribute__((ext_vector_type(2))) float v2f;
typedef __attribute__((ext_vector_type(8))) float v8f;

constexpr int Bc  = 512;     // batch
constexpr int Tc  = 256;     // time steps
constexpr int Fc  = 128;     // input features
constexpr int Hc  = 100;     // LSTM units
constexpr int G4  = 400;     // 4*H
constexpr int BT  = Bc * Tc; // 131072 rows
constexpr int HP  = 101;     // padded h stride in LDS (odd -> bank friendly)

// ---------------------------------------------------------------------------
// GEMM: C[M,400] = A[M,K] @ W[K,400] using v_wmma_f32_16x16x4_f32.
// One wave computes a 16x80 output strip (5 N-tiles). 8 waves per block.
// ---------------------------------------------------------------------------
__global__ __launch_bounds__(256) void gemm_wmma_f32(
    const float* __restrict__ A, const float* __restrict__ W,
    float* __restrict__ C, int M, int K)
{
  const int wave = blockIdx.x * 8 + (threadIdx.x >> 5);
  const int lane = threadIdx.x & 31;
  const int mt = wave / 5;          // M tile index (16 rows)
  const int ng = wave % 5;          // N group (80 cols)
  if (mt * 16 >= M) return;
  const int m0 = mt * 16;
  const int n0 = ng * 80;
  const int nn = lane & 15;         // N-within-tile (B/C lanes) / M row (A lanes)
  const int kk = (lane >> 4) * 2;   // K sub-offset 0 or 2
  const int mrow_hi = (lane >> 4) * 8;

  v8f acc[5] = {};
  for (int k = 0; k < K; k += 4) {
    // A fragment: lane (nn) = row, regs hold K = k+kk, k+kk+1 (8B aligned)
    v2f a = *(const v2f*)(A + (size_t)(m0 + nn) * K + k + kk);
#pragma unroll
    for (int nt = 0; nt < 5; ++nt) {
      v2f b;
      b.x = W[(size_t)(k + kk) * G4 + n0 + nt * 16 + nn];
      b.y = W[(size_t)(k + kk + 1) * G4 + n0 + nt * 16 + nn];
      acc[nt] = __builtin_amdgcn_wmma_f32_16x16x4_f32(
          false, a, false, b, (short)0, acc[nt], false, false);
    }
  }
#pragma unroll
  for (int nt = 0; nt < 5; ++nt)
#pragma unroll
    for (int r = 0; r < 8; ++r)
      C[(size_t)(m0 + r + mrow_hi) * G4 + n0 + nt * 16 + nn] = acc[nt][r];
}

// ---------------------------------------------------------------------------
// LSTM recurrence for one layer. One workgroup = 16 batch rows, 25 waves,
// wave w owns z columns [16w, 16w+16). U fragments preloaded in registers.
// XW = precomputed input projection [B*T, 400] (row = b*T + t).
// ---------------------------------------------------------------------------
template <bool WRITE_SEQ>
__global__ __launch_bounds__(800) void lstm_layer(
    const float* __restrict__ XW, const float* __restrict__ U,
    float* __restrict__ seq_out,   // [B*T, 100] if WRITE_SEQ
    float* __restrict__ h_last)    // [B, 100]   if !WRITE_SEQ
{
  const int wave = threadIdx.x >> 5;   // 0..24
  const int lane = threadIdx.x & 31;
  const int b0   = blockIdx.x * 16;

  __shared__ float hlds[16 * HP];      // h_{t} state, row-major, stride 101
  __shared__ float zlds[16 * G4];      // activated gate values

  const int n0 = wave * 16;
  const int nn = lane & 15;
  const int kk = (lane >> 4) * 2;
  const int mrow_hi = (lane >> 4) * 8;

  // Preload this wave's U column stripe as 25 B-fragments (lives in VGPRs).
  v2f ufrag[25];
#pragma unroll
  for (int kc = 0; kc < 25; ++kc) {
    const int kr = kc * 4 + kk;
    ufrag[kc].x = U[(size_t)kr * G4 + n0 + nn];
    ufrag[kc].y = U[(size_t)(kr + 1) * G4 + n0 + nn];
  }

  // Gate type for this lane's z column (constant over time):
  // cols [0,100)=i, [100,200)=f sigmoid; [200,300)=g tanh; [300,400)=o sigmoid
  const int zcol = n0 + nn;
  const float gs = (zcol >= 200 && zcol < 300) ? 2.0f : 1.0f;  // act(x)=gs*sig(gs*x)-(gs-1)

  // Each lane owns 2 of the 16x100 cells; c state in registers.
  const int idx0 = (wave * 32 + lane) * 2;       // 0..1598
  const int cb0 = idx0 / Hc,       cj0 = idx0 % Hc;
  const int cb1 = (idx0 + 1) / Hc, cj1 = (idx0 + 1) % Hc;
  float cst0 = 0.0f, cst1 = 0.0f;

  for (int i = threadIdx.x; i < 16 * HP; i += 800) hlds[i] = 0.0f;
  __syncthreads();

  for (int t = 0; t < Tc; ++t) {
    // C initializer = precomputed x_t @ W for this tile
    v8f acc;
#pragma unroll
    for (int r = 0; r < 8; ++r)
      acc[r] = XW[((size_t)(b0 + r + mrow_hi) * Tc + t) * G4 + zcol];

    // z += h_{t-1} @ U  (25 K-steps of 4, A from LDS, B from registers)
#pragma unroll
    for (int kc = 0; kc < 25; ++kc) {
      const int kr = kc * 4 + kk;
      v2f a;
      a.x = hlds[nn * HP + kr];
      a.y = hlds[nn * HP + kr + 1];
      acc = __builtin_amdgcn_wmma_f32_16x16x4_f32(
          false, a, false, ufrag[kc], (short)0, acc, false, false);
    }

    // Apply gate nonlinearity and publish to LDS
#pragma unroll
    for (int r = 0; r < 8; ++r) {
      const float v = gs / (1.0f + __expf(-gs * acc[r])) - (gs - 1.0f);
      zlds[(r + mrow_hi) * G4 + zcol] = v;
    }
    __syncthreads();   // z complete; all h_{t-1} reads complete

    // Elementwise state update: c = f*c + i*g ; h = o * tanh(c)
    {
      const float* z = zlds + cb0 * G4 + cj0;
      const float ig = z[0], fg = z[100], gg = z[200], og = z[300];
      cst0 = fg * cst0 + ig * gg;
      const float h = og * (2.0f / (1.0f + __expf(-2.0f * cst0)) - 1.0f);
      hlds[cb0 * HP + cj0] = h;
      if (WRITE_SEQ) seq_out[((size_t)(b0 + cb0) * Tc + t) * Hc + cj0] = h;
      else if (t == Tc - 1) h_last[(size_t)(b0 + cb0) * Hc + cj0] = h;
    }
    {
      const float* z = zlds + cb1 * G4 + cj1;
      const float ig = z[0], fg = z[100], gg = z[200], og = z[300];
      cst1 = fg * cst1 + ig * gg;
      const float h = og * (2.0f / (1.0f + __expf(-2.0f * cst1)) - 1.0f);
      hlds[cb1 * HP + cj1] = h;
      if (WRITE_SEQ) seq_out[((size_t)(b0 + cb1) * Tc + t) * Hc + cj1] = h;
      else if (t == Tc - 1) h_last[(size_t)(b0 + cb1) * Hc + cj1] = h;
    }
    __syncthreads();   // h_t visible before next step's A loads
  }
}

// ---------------------------------------------------------------------------
// Dense head: out[b] = h_last[b,:] @ Wd + bd
// ---------------------------------------------------------------------------
__global__ void dense_head(const float* __restrict__ h,
                           const float* __restrict__ Wd,
                           const float* __restrict__ bd,
                           float* __restrict__ out)
{
  const int b = blockIdx.x * blockDim.x + threadIdx.x;
  if (b < Bc) {
    float s = bd[0];
    for (int j = 0; j < Hc; ++j) s = fmaf(h[b * Hc + j], Wd[j], s);
    out[b] = s;
  }
}

// ---------------------------------------------------------------------------
extern "C" void kernel_launch(void* const* d_in, const int* in_sizes, int n_in,
                              void* d_out, int out_size, void* d_ws, size_t ws_size,
                              hipStream_t stream) {
  const float* x  = (const float*)d_in[0];
  const float* W1 = (const float*)d_in[1];
  const float* U1 = (const float*)d_in[2];
  const float* W2 = (const float*)d_in[3];
  const float* U2 = (const float*)d_in[4];
  const float* Wd = (const float*)d_in[5];
  const float* bd = (const float*)d_in[6];
  float* out = (float*)d_out;

  char* ws = (char*)d_ws;
  float* XW    = (float*)ws;                                   // [B*T, 400]
  float* seq1  = (float*)(ws + (size_t)BT * G4 * sizeof(float));      // [B*T, 100]
  float* hlast = (float*)(ws + (size_t)BT * (G4 + Hc) * sizeof(float)); // [B, 100]

  const int gemmBlocks = (BT / 16) * 5 / 8;  // 8 waves per 256-thread block

  // Layer 1: input projection, then recurrence (writes full sequence)
  gemm_wmma_f32<<<gemmBlocks, 256, 0, stream>>>(x, W1, XW, BT, Fc);
  lstm_layer<true><<<Bc / 16, 800, 0, stream>>>(XW, U1, seq1, nullptr);

  // Layer 2: projection of the hidden sequence, then recurrence (keep last h)
  gemm_wmma_f32<<<gemmBlocks, 256, 0, stream>>>(seq1, W2, XW, BT, Hc);
  lstm_layer<false><<<Bc / 16, 800, 0, stream>>>(XW, U2, nullptr, hlast);

  dense_head<<<(Bc + 255) / 256, 256, 0, stream>>>(hlast, Wd, bd, out);
}